// CG_base_py_3813930959604
// MI455X (gfx1250) — compile-verified
//
#include <hip/hip_runtime.h>

typedef __attribute__((ext_vector_type(2))) float v2f;
typedef __attribute__((ext_vector_type(8))) float v8f;

namespace {
constexpr int kLmax   = 5;
constexpr int kTau    = 24;
constexpr int kBatch  = 128;
constexpr int kNT     = 69;
constexpr int kNcols  = kTau * kTau;   // 576
constexpr int kNTiles = kNcols / 16;   // 36 N-tiles of 16 columns
constexpr int kKmaxPad = 124;          // ceil(121/4)*4

struct Meta {
  short l1[kNT], l2[kNT], l[kNT], slot[kNT];
  int   nl[6];
  int   outBase[6];   // float offset of out[l] within d_out
  unsigned magic[6];  // ceil(2^16 / (2l+1)) : exact i/pdim for i < 512
};

constexpr Meta buildMeta() {
  Meta m{};
  int cnt[6] = {0, 0, 0, 0, 0, 0};
  int i = 0;
  for (int a = 0; a <= kLmax; ++a)
    for (int b = 0; b <= a; ++b) {
      int hi = (a + b < kLmax) ? (a + b) : kLmax;
      for (int c = a - b; c <= hi; ++c) {
        m.l1[i] = (short)a; m.l2[i] = (short)b; m.l[i] = (short)c;
        m.slot[i] = (short)cnt[c];
        ++cnt[c]; ++i;
      }
    }
  int base = 0;
  for (int c = 0; c < 6; ++c) {
    m.nl[c] = cnt[c];
    m.outBase[c] = base;
    base += kBatch * cnt[c] * kNcols * (2 * c + 1) * 2;
    int d = 2 * c + 1;
    m.magic[c] = (unsigned)((65536 + d - 1) / d);
  }
  return m;
}
} // namespace

__device__ __constant__ Meta c_meta = buildMeta();

// One block per (triple, batch), 256 threads = 8 waves.
// Each wave owns N-tiles of 16 columns; per tile it runs TWO independent
// V_WMMA_F32_16X16X4_F32 accumulator chains (real & imag) sharing one
// A fragment (CG rows) and one set of complex-operand LDS loads.
// D tiles are staged in per-wave LDS (b128 stores) and streamed out as
// contiguous, coalesced b64 stores (dominant cost: 269 MB output stream).
__global__ __launch_bounds__(256) void cg_tensor_product(
    const float* __restrict__ f0, const float* __restrict__ f1,
    const float* __restrict__ f2, const float* __restrict__ f3,
    const float* __restrict__ f4, const float* __restrict__ f5,
    const float* __restrict__ cg, float* __restrict__ out) {
  const int tri = blockIdx.x;
  const int b   = blockIdx.y;

  const int l1   = c_meta.l1[tri];
  const int l2   = c_meta.l2[tri];
  const int lo   = c_meta.l[tri];
  const int slot = c_meta.slot[tri];
  const int m1   = 2 * l1 + 1;
  const int m2   = 2 * l2 + 1;
  const int pdim = 2 * lo + 1;
  const int K    = m1 * m2;
  const int Kpad = (K + 3) & ~3;
  const int nl      = c_meta.nl[lo];
  const int outBase = c_meta.outBase[lo];
  const unsigned mgc = c_meta.magic[lo];

  const float* fs[6] = {f0, f1, f2, f3, f4, f5};
  const float* fa = fs[l1];
  const float* fb = fs[l2];

  __shared__ float2 s_a[11 * kTau];          // (re,im) of a[m][t]
  __shared__ float2 s_b[11 * kTau];          // (re,im) of b[n][u]
  __shared__ float  s_cg[16 * kKmaxPad];     // [row p][k], zero-padded
  __shared__ uint2  s_idx2[kKmaxPad / 2];    // pairs of (m*24)<<16 | (n*24)
  __shared__ float2 s_stage[8][16 * 16];     // per-wave D-tile staging

  // Stage a = f_{l1}[b], b = f_{l2}[b] into LDS as [m*24+t] float2 (re,im).
  // Input layout: f_l is (B, 24, 2l+1, 2): elem (b,t,m,ri) at ((b*24+t)*m1+m)*2+ri.
  for (int i = threadIdx.x; i < m1 * kTau; i += 256) {
    int m = i / kTau, t = i - m * kTau;
    s_a[i] = *(const float2*)(fa + (size_t)((b * kTau + t) * m1 + m) * 2);
  }
  for (int i = threadIdx.x; i < m2 * kTau; i += 256) {
    int m = i / kTau, t = i - m * kTau;
    s_b[i] = *(const float2*)(fb + (size_t)((b * kTau + t) * m2 + m) * 2);
  }
  // CG tile, zero-padded to 16 rows x Kpad cols. cg_all is (69, 11, 121).
  for (int i = threadIdx.x; i < 16 * Kpad; i += 256) {
    int row = i / Kpad, k = i - row * Kpad;
    float v = 0.0f;
    if (row < pdim && k < K) v = cg[(size_t)(tri * 11 + row) * 121 + k];
    s_cg[row * Kpad + k] = v;
  }
  // k -> (m,n) decode table, pre-scaled by 24 for direct LDS indexing.
  {
    unsigned* idxu = (unsigned*)s_idx2;
    for (int k = threadIdx.x; k < Kpad; k += 256) {
      unsigned v = 0u;
      if (k < K) {
        int m = k / m2, nn = k - m * m2;
        v = ((unsigned)(m * kTau) << 16) | (unsigned)(nn * kTau);
      }
      idxu[k] = v;
    }
  }
  __syncthreads();

  const int wave = threadIdx.x >> 5;
  const int lane = threadIdx.x & 31;
  const int half = lane >> 4;   // lane group 0/1
  const int lrow = lane & 15;
  float2* stage = s_stage[wave];

  for (int ntile = wave; ntile < kNTiles; ntile += 8) {  // wave-uniform
    const int col = ntile * 16 + lrow;   // column (t,u) for B and D
    const int t   = col / kTau;
    const int u   = col - t * kTau;

    v8f accr = {0.f, 0.f, 0.f, 0.f, 0.f, 0.f, 0.f, 0.f};
    v8f acci = {0.f, 0.f, 0.f, 0.f, 0.f, 0.f, 0.f, 0.f};
    const int cgRow = lrow * Kpad;

    for (int kk = 0; kk < Kpad; kk += 4) {
      // A fragment (16x4 f32): lanes 0-15 hold K=kk,kk+1 in v0,v1;
      // lanes 16-31 hold K=kk+2,kk+3 (ISA 16x4 A layout).
      const int ka = kk + half * 2;
      float2 cgv = *(const float2*)&s_cg[cgRow + ka];   // 8B-aligned
      v2f A; A.x = cgv.x; A.y = cgv.y;

      // B fragments (4x16): lane = N; k rows split across lane halves
      // symmetric to A. Build complex outer-product values on the fly.
      uint2 ii = s_idx2[ka >> 1];
      int ma0 = (int)(ii.x >> 16), nb0 = (int)(ii.x & 0xffffu);
      int ma1 = (int)(ii.y >> 16), nb1 = (int)(ii.y & 0xffffu);
      float2 a0 = s_a[ma0 + t], bb0 = s_b[nb0 + u];
      float2 a1 = s_a[ma1 + t], bb1 = s_b[nb1 + u];
      v2f Br, Bi;
      Br.x = a0.x * bb0.x - a0.y * bb0.y;
      Bi.x = a0.x * bb0.y + a0.y * bb0.x;
      Br.y = a1.x * bb1.x - a1.y * bb1.y;
      Bi.y = a1.x * bb1.y + a1.y * bb1.x;

      accr = __builtin_amdgcn_wmma_f32_16x16x4_f32(
          false, A, false, Br, (short)0, accr, false, false);
      acci = __builtin_amdgcn_wmma_f32_16x16x4_f32(
          false, A, false, Bi, (short)0, acci, false, false);
    }

    // Stage D (re,im interleaved) unconditionally: [col_local][prow 0..15].
    // D layout: VGPR r -> M = r + 8*half, N = lrow.
#pragma unroll
    for (int r = 0; r < 8; ++r) {
      int prow = r + half * 8;
      float2 v; v.x = accr[r]; v.y = acci[r];
      stage[lrow * 16 + prow] = v;
    }

    // The 16-column span [b, slot*576 + ntile*16 .. +16, :, :] is contiguous:
    // 16*pdim float2. Stream it with fully coalesced b64 stores.
    // i/pdim via exact magic multiply (pdim in {1,3,5,7,9,11}, i < 512).
    const int spanF2 = 16 * pdim;
    float2* gout = (float2*)out + (size_t)(outBase >> 1) +
                   (size_t)((b * nl + slot) * kNcols + ntile * 16) * pdim;
    for (int i = lane; i < spanF2; i += 32) {
      int c = (int)(((unsigned)i * mgc) >> 16);   // col_local = i / pdim
      int p = i - c * pdim;                       // prow      = i % pdim
      gout[i] = stage[c * 16 + p];
    }
  }
}

extern "C" void kernel_launch(void* const* d_in, const int* in_sizes, int n_in,
                              void* d_out, int out_size, void* d_ws,
                              size_t ws_size, hipStream_t stream) {
  const float* f0 = (const float*)d_in[0];
  const float* f1 = (const float*)d_in[1];
  const float* f2 = (const float*)d_in[2];
  const float* f3 = (const float*)d_in[3];
  const float* f4 = (const float*)d_in[4];
  const float* f5 = (const float*)d_in[5];
  const float* cg = (const float*)d_in[6];
  float* out = (float*)d_out;

  dim3 grid(kNT, kBatch);
  cg_tensor_product<<<grid, 256, 0, stream>>>(f0, f1, f2, f3, f4, f5, cg, out);
}